// HyperFNS_55645596286993
// MI455X (gfx1250) — compile-verified
//
#include <hip/hip_runtime.h>
#include <hip/hip_bf16.h>
#include <math.h>

// ---------------------------------------------------------------------------
// Problem constants
// ---------------------------------------------------------------------------
#define GN      255              // PDE grid size
#define NN      (GN * GN)        // 65025
#define PN      256              // padded spectral grid (16x16 tiles, no guards)
#define PNN     (PN * PN)        // 65536
#define NB      16               // batch
#define BNN     ((size_t)NB * NN)
#define BPNN    ((size_t)NB * PNN)
#define MIDC    32               // mid channels of ConvT chain
#define KOUT    3                // outer iterations
#define NITERS  10               // Chebyshev steps
#define ALPHA   3.0f

typedef __attribute__((ext_vector_type(2))) float v2f;
typedef __attribute__((ext_vector_type(8))) float v8f;

// ---------------------------------------------------------------------------
// Pseudo-random init for power iteration (jax RNG is not bit-reproducible
// anyway; power iteration converges to the same spectral bound).
// ---------------------------------------------------------------------------
__global__ void init_u_kernel(float* __restrict__ u, unsigned seed) {
  size_t idx = (size_t)blockIdx.x * blockDim.x + threadIdx.x;
  if (idx >= BNN) return;
  unsigned h = (unsigned)idx * 2654435761u ^ (seed * 0x9E3779B9u + 0x7F4A7C15u);
  h ^= h >> 16; h *= 0x85EBCA6Bu; h ^= h >> 13; h *= 0xC2B2AE35u; h ^= h >> 16;
  u[idx] = (float)h * (2.0f / 4294967296.0f) - 1.0f;
}

// ---------------------------------------------------------------------------
// Per-sample 3x3 "grouped" conv, zero padding (SAME).
// ---------------------------------------------------------------------------
__device__ __forceinline__ float conv3x3_at(const float* __restrict__ xb,
                                            const float* __restrict__ k9,
                                            int py, int px) {
  float acc = 0.0f;
#pragma unroll
  for (int ky = 0; ky < 3; ky++) {
    int iy = py + ky - 1;
    if (iy < 0 || iy >= GN) continue;
#pragma unroll
    for (int kx = 0; kx < 3; kx++) {
      int ix = px + kx - 1;
      if (ix < 0 || ix >= GN) continue;
      acc += k9[ky * 3 + kx] * xb[iy * GN + ix];
    }
  }
  return acc;
}

__global__ void conv3x3_kernel(const float* __restrict__ x,
                               const float* __restrict__ kA,
                               float* __restrict__ y) {
  size_t idx = (size_t)blockIdx.x * blockDim.x + threadIdx.x;
  if (idx >= BNN) return;
  int b  = (int)(idx / NN);
  int p  = (int)(idx % NN);
  y[idx] = conv3x3_at(x + (size_t)b * NN, kA + b * 9, p / GN, p % GN);
}

// r = f - conv(x)  (unpadded, for the final norm)
__global__ void residual_kernel(const float* __restrict__ x,
                                const float* __restrict__ f,
                                const float* __restrict__ kA,
                                float* __restrict__ r) {
  size_t idx = (size_t)blockIdx.x * blockDim.x + threadIdx.x;
  if (idx >= BNN) return;
  int b = (int)(idx / NN);
  int p = (int)(idx % NN);
  r[idx] = f[idx] - conv3x3_at(x + (size_t)b * NN, kA + b * 9, p / GN, p % GN);
}

// r = f - conv(x) written into the zero-padded 256x256 spectral layout
__global__ void residual_pad_kernel(const float* __restrict__ x,
                                    const float* __restrict__ f,
                                    const float* __restrict__ kA,
                                    float* __restrict__ rp) {
  size_t idx = (size_t)blockIdx.x * blockDim.x + threadIdx.x;
  if (idx >= BPNN) return;
  int b = (int)(idx / PNN);
  int p = (int)(idx % PNN);
  int y = p / PN, xx = p % PN;
  if (y >= GN || xx >= GN) { rp[idx] = 0.0f; return; }
  size_t q = (size_t)b * NN + (size_t)y * GN + xx;
  rp[idx] = f[q] - conv3x3_at(x + (size_t)b * NN, kA + b * 9, y, xx);
}

// xout = xin + tau(m,root) * (f - conv(xin))
__global__ void cheby_step_kernel(const float* __restrict__ xin,
                                  float* __restrict__ xout,
                                  const float* __restrict__ f,
                                  const float* __restrict__ kA,
                                  const float* __restrict__ m,
                                  float root) {
  size_t idx = (size_t)blockIdx.x * blockDim.x + threadIdx.x;
  if (idx >= BNN) return;
  int b = (int)(idx / NN);
  int p = (int)(idx % NN);
  float mb    = m[b];
  float denom = mb + mb / ALPHA - (mb / ALPHA - mb) * root;
  float tau   = 2.0f / denom;
  float cv = conv3x3_at(xin + (size_t)b * NN, kA + b * 9, p / GN, p % GN);
  xout[idx] = xin[idx] + tau * (f[idx] - cv);
}

// per-sample max |y| -> m[b]   (one block per sample; LDS tree reduction)
__global__ void rowmax_abs_kernel(const float* __restrict__ y,
                                  float* __restrict__ m) {
  __shared__ float sm[256];
  int b = blockIdx.x;
  const float* p = y + (size_t)b * NN;
  float mx = 0.0f;
  for (int i = threadIdx.x; i < NN; i += 256) mx = fmaxf(mx, fabsf(p[i]));
  sm[threadIdx.x] = mx;
  __syncthreads();
  for (int s = 128; s > 0; s >>= 1) {
    if ((int)threadIdx.x < s) sm[threadIdx.x] = fmaxf(sm[threadIdx.x], sm[threadIdx.x + s]);
    __syncthreads();
  }
  if (threadIdx.x == 0) m[b] = sm[0];
}

__global__ void scale_kernel(const float* __restrict__ y,
                             const float* __restrict__ m,
                             float* __restrict__ u) {
  size_t idx = (size_t)blockIdx.x * blockDim.x + threadIdx.x;
  if (idx >= BNN) return;
  u[idx] = y[idx] / m[idx / NN];
}

// x (255x255 layout) += e (padded 256x256 layout)
__global__ void add_from_pad_kernel(float* __restrict__ x,
                                    const float* __restrict__ ep) {
  size_t idx = (size_t)blockIdx.x * blockDim.x + threadIdx.x;
  if (idx >= BNN) return;
  int b = (int)(idx / NN);
  int p = (int)(idx % NN);
  int y = p / GN, xx = p % GN;
  x[idx] += ep[(size_t)b * PNN + (size_t)y * PN + xx];
}

// ---------------------------------------------------------------------------
// ConvTranspose2d (gather form) + optional exact GELU.
// PyTorch weight layout [Cin, Cout, 3, 3].
// ---------------------------------------------------------------------------
__global__ void convt_kernel(const float* __restrict__ in,
                             const float* __restrict__ w,
                             const float* __restrict__ bias,
                             float* __restrict__ out,
                             int Cin, int Cout, int inHW, int outHW,
                             int stride, int pad, int doGelu) {
  size_t idx = (size_t)blockIdx.x * blockDim.x + threadIdx.x;
  size_t total = (size_t)NB * Cout * outHW * outHW;
  if (idx >= total) return;
  int ox = (int)(idx % outHW); size_t t = idx / outHW;
  int oy = (int)(t % outHW);   t /= outHW;
  int co = (int)(t % Cout);
  int b  = (int)(t / Cout);
  float acc = bias[co];
#pragma unroll
  for (int ky = 0; ky < 3; ky++) {
    int ty = oy + pad - ky;
    if (ty < 0 || (ty % stride) != 0) continue;
    int iy = ty / stride;
    if (iy >= inHW) continue;
#pragma unroll
    for (int kx = 0; kx < 3; kx++) {
      int tx = ox + pad - kx;
      if (tx < 0 || (tx % stride) != 0) continue;
      int ix = tx / stride;
      if (ix >= inHW) continue;
      const float* ip = in + (((size_t)b * Cin) * inHW + iy) * inHW + ix;
      const float* wp = w + co * 9 + ky * 3 + kx;
      for (int ci = 0; ci < Cin; ci++)
        acc += wp[(size_t)ci * Cout * 9] * ip[(size_t)ci * inHW * inHW];
    }
  }
  if (doGelu) acc = 0.5f * acc * (1.0f + erff(acc * 0.70710678118654752f));
  out[idx] = acc;
}

// torch .view(B,1,N,N,2) reinterpretation of contiguous [B,2,N,N], written
// into the padded spectral layout (pads zeroed).
__global__ void extract_weights_pad_kernel(const float* __restrict__ h,
                                           float* __restrict__ wRe,
                                           float* __restrict__ wIm) {
  size_t idx = (size_t)blockIdx.x * blockDim.x + threadIdx.x;
  if (idx >= BPNN) return;
  int b = (int)(idx / PNN);
  int p = (int)(idx % PNN);
  int y = p / PN, x = p % PN;
  if (y >= GN || x >= GN) { wRe[idx] = 0.0f; wIm[idx] = 0.0f; return; }
  size_t src = (size_t)b * 2 * NN + (size_t)y * 2 * GN + (size_t)x * 2;
  wRe[idx] = h[src];
  wIm[idx] = h[src + 1];
}

// ---------------------------------------------------------------------------
// DFT twiddle matrices (ortho normalized), padded to 256x256 with zeros.
// F = exp(-2pi i jk/255)/sqrt(255), Finv = conj(F).
// ---------------------------------------------------------------------------
__global__ void gen_dft_pad_kernel(float* __restrict__ Fre, float* __restrict__ Fim,
                                   float* __restrict__ FiRe, float* __restrict__ FiIm) {
  int idx = blockIdx.x * blockDim.x + threadIdx.x;
  if (idx >= PNN) return;
  int j = idx / PN, k = idx % PN;
  if (j >= GN || k >= GN) {
    Fre[idx] = 0.0f; Fim[idx] = 0.0f; FiRe[idx] = 0.0f; FiIm[idx] = 0.0f;
    return;
  }
  int tw = (j * k) % GN;
  float ang = -6.2831853071795864769f * (float)tw / (float)GN;
  float s, c;
  sincosf(ang, &s, &c);
  const float inv = 0.062622429108514952f;  // 1/sqrt(255)
  Fre[idx] = c * inv;  Fim[idx] = s * inv;
  FiRe[idx] = c * inv; FiIm[idx] = -s * inv;
}

// ---------------------------------------------------------------------------
// Complex GEMM on the padded 256x256x256 domain using V_WMMA_F32_16X16X4_F32.
// One wave owns one full 16x16 tile: no bounds checks, no EXEC manipulation,
// contiguous b64 A-fragment loads. Zero pads self-propagate through the GEMM.
//   ACPLX : A has an imaginary part       (compile-time)
//   OUTIM : imaginary output is written   (compile-time)
// Three accumulators avoid per-iteration negation:
//   re = acc(ar*br) - acc(ai*bi),  im = acc(ar*bi) + acc(ai*br)
// A 16x4 layout: lanes 0-15 hold (K,K+1) of row M=lane; lanes 16-31 (K+2,K+3).
// C/D: VGPR v -> row tr*16 + v + 8*half, col = tc*16 + (lane&15).
// ---------------------------------------------------------------------------
template <bool ACPLX, bool OUTIM>
__global__ void __launch_bounds__(256)
cgemm256_wmma_kernel(const float* __restrict__ aRe, const float* __restrict__ aIm,
                     long aStride,
                     const float* __restrict__ bRe, const float* __restrict__ bIm,
                     long bStride,
                     float* __restrict__ oRe, float* __restrict__ oIm) {
  int lane = threadIdx.x & 31;
  int wv   = threadIdx.x >> 5;
  int tile = blockIdx.x * 8 + wv;     // 16 samples * 256 tiles = 4096
  int b    = tile >> 8;
  int tr   = (tile >> 4) & 15;
  int tc   = tile & 15;
  int half = lane >> 4;               // 0: K+0/1, 1: K+2/3
  int lid  = lane & 15;
  int arow = tr * 16 + lid;
  int bcol = tc * 16 + lid;

  const float* aR = aRe + (size_t)b * aStride + (size_t)arow * PN;
  const float* aI = ACPLX ? (aIm + (size_t)b * aStride + (size_t)arow * PN) : nullptr;
  const float* bR = bRe + (size_t)b * bStride + bcol;
  const float* bI = bIm + (size_t)b * bStride + bcol;

  v8f c_rr = {};   // acc(ar*br)
  v8f c_ii = {};   // acc(ai*bi)
  v8f c_im = {};   // acc(ar*bi) + acc(ai*br)

#pragma unroll 4
  for (int k0 = 0; k0 < PN; k0 += 4) {
    int ka = k0 + 2 * half;
    if (k0 + 8 < PN)
      __builtin_prefetch(bR + (size_t)(ka + 8) * PN, 0, 1);

    v2f ar = *(const v2f*)(aR + ka);
    v2f br, bi;
    br.x = bR[(size_t)ka * PN];
    br.y = bR[(size_t)(ka + 1) * PN];
    bi.x = bI[(size_t)ka * PN];
    bi.y = bI[(size_t)(ka + 1) * PN];

    c_rr = __builtin_amdgcn_wmma_f32_16x16x4_f32(false, ar, false, br,
                                                 (short)0, c_rr, false, false);
    c_im = __builtin_amdgcn_wmma_f32_16x16x4_f32(false, ar, false, bi,
                                                 (short)0, c_im, false, false);
    if (ACPLX) {
      v2f ai = *(const v2f*)(aI + ka);
      c_ii = __builtin_amdgcn_wmma_f32_16x16x4_f32(false, ai, false, bi,
                                                   (short)0, c_ii, false, false);
      c_im = __builtin_amdgcn_wmma_f32_16x16x4_f32(false, ai, false, br,
                                                   (short)0, c_im, false, false);
    }
  }

  size_t obase = (size_t)b * PNN + (size_t)(tr * 16 + half * 8) * PN + bcol;
  float* oR = oRe + obase;
#pragma unroll
  for (int v = 0; v < 8; v++) {
    float re = ACPLX ? (c_rr[v] - c_ii[v]) : c_rr[v];
    oR[(size_t)v * PN] = re;
  }
  if (OUTIM) {
    float* oI = oIm + obase;
#pragma unroll
    for (int v = 0; v < 8; v++) oI[(size_t)v * PN] = c_im[v];
  }
}

// ---------------------------------------------------------------------------
// S = HermitianExtend( rhat .* W ) on the padded layout: cols [0,128) direct;
// cols v in [128,255): S[u][v] = conj( (rhat.*W)[(255-u)%255][255-v] ).
// Pad row/col written as zero.
// ---------------------------------------------------------------------------
__global__ void spec_mul_herm_kernel(const float* __restrict__ hRe,
                                     const float* __restrict__ hIm,
                                     const float* __restrict__ wRe,
                                     const float* __restrict__ wIm,
                                     float* __restrict__ sRe,
                                     float* __restrict__ sIm) {
  size_t idx = (size_t)blockIdx.x * blockDim.x + threadIdx.x;
  if (idx >= BPNN) return;
  int b = (int)(idx / PNN);
  int p = (int)(idx % PNN);
  int u = p / PN, v = p % PN;
  if (u >= GN || v >= GN) { sRe[idx] = 0.0f; sIm[idx] = 0.0f; return; }
  int uu = u, vv = v;
  float sign = 1.0f;
  const int halfc = GN / 2 + 1;  // 128 kept columns (rfft truncation)
  if (v >= halfc) { uu = (GN - u) % GN; vv = GN - v; sign = -1.0f; }
  size_t q = (size_t)b * PNN + (size_t)uu * PN + vv;
  float ar = hRe[q], ai = hIm[q];
  float br = wRe[q], bi = wIm[q];
  sRe[idx] = ar * br - ai * bi;
  sIm[idx] = sign * (ar * bi + ai * br);
}

// ---------------------------------------------------------------------------
// Final Frobenius norm.
// ---------------------------------------------------------------------------
__global__ void zero1_kernel(float* p) {
  if (blockIdx.x == 0 && threadIdx.x == 0) *p = 0.0f;
}

__global__ void sumsq_kernel(const float* __restrict__ r, float* __restrict__ acc) {
  __shared__ float sm[256];
  float s = 0.0f;
  for (size_t i = (size_t)blockIdx.x * 256 + threadIdx.x; i < BNN;
       i += (size_t)gridDim.x * 256)
    s += r[i] * r[i];
  sm[threadIdx.x] = s;
  __syncthreads();
  for (int t = 128; t > 0; t >>= 1) {
    if ((int)threadIdx.x < t) sm[threadIdx.x] += sm[threadIdx.x + t];
    __syncthreads();
  }
  if (threadIdx.x == 0) atomicAdd(acc, sm[0]);
}

__global__ void finish_norm_kernel(const float* acc, float* out) {
  if (blockIdx.x == 0 && threadIdx.x == 0) out[0] = sqrtf(*acc);
}

// ---------------------------------------------------------------------------
// Host orchestration
// ---------------------------------------------------------------------------
extern "C" void kernel_launch(void* const* d_in, const int* in_sizes, int n_in,
                              void* d_out, int out_size, void* d_ws, size_t ws_size,
                              hipStream_t stream) {
  const float* x0 = (const float*)d_in[0];
  const float* f  = (const float*)d_in[1];
  const float* kA = (const float*)d_in[2];
  const float* Wt[7] = {(const float*)d_in[3],  (const float*)d_in[5],
                        (const float*)d_in[7],  (const float*)d_in[9],
                        (const float*)d_in[11], (const float*)d_in[13],
                        (const float*)d_in[15]};
  const float* Bs[7] = {(const float*)d_in[4],  (const float*)d_in[6],
                        (const float*)d_in[8],  (const float*)d_in[10],
                        (const float*)d_in[12], (const float*)d_in[14],
                        (const float*)d_in[16]};

  // ---- workspace layout (floats; every block even-sized -> 8B alignment) ----
  float* W = (float*)d_ws;
  size_t off = 0;
  auto alloc = [&](size_t n) { float* p = W + off; off += (n + 1) & ~(size_t)1; return p; };
  float* xA   = alloc(BNN);
  float* xB   = alloc(BNN);
  float* ubuf = alloc(BNN);
  float* ybuf = alloc(BNN);
  float* rbuf = alloc(BNN);
  float* mbuf = alloc(16);
  const size_t CTMAX = (size_t)NB * MIDC * 129 * 129;
  float* ct0  = alloc(CTMAX);
  float* ct1  = alloc(CTMAX);
  float* hbuf = alloc((size_t)NB * 2 * NN);
  float* wRe  = alloc(BPNN);
  float* wIm  = alloc(BPNN);
  float* Fre  = alloc(PNN);
  float* Fim  = alloc(PNN);
  float* FiRe = alloc(PNN);
  float* FiIm = alloc(PNN);
  float* rp   = alloc(BPNN);
  float* ep   = alloc(BPNN);
  float* t1Re = alloc(BPNN);
  float* t1Im = alloc(BPNN);
  float* t2Re = alloc(BPNN);
  float* t2Im = alloc(BPNN);
  float* sRe  = alloc(BPNN);
  float* sIm  = alloc(BPNN);
  float* nacc = alloc(2);
  (void)in_sizes; (void)n_in; (void)out_size; (void)ws_size;

  const int TB    = 256;
  const int gBNN  = (int)((BNN + TB - 1) / TB);
  const int gBPNN = (int)((BPNN + TB - 1) / TB);

  // ---- DFT matrices (padded) ----
  gen_dft_pad_kernel<<<(PNN + TB - 1) / TB, TB, 0, stream>>>(Fre, Fim, FiRe, FiIm);

  // ---- spectral weights (kernelA-only: hoisted out of the K loop) ----
  {
    int sz[8] = {3, 5, 9, 17, 33, 65, 129, 255};
    const float* cin = kA;
    float* bufs[2] = {ct0, ct1};
    for (int L = 0; L < 6; L++) {
      int Cin = (L == 0) ? 1 : MIDC;
      float* out = bufs[L & 1];
      size_t tot = (size_t)NB * MIDC * sz[L + 1] * sz[L + 1];
      convt_kernel<<<(int)((tot + TB - 1) / TB), TB, 0, stream>>>(
          cin, Wt[L], Bs[L], out, Cin, MIDC, sz[L], sz[L + 1], 2, 1, 1);
      cin = out;
    }
    size_t tot = (size_t)NB * 2 * NN;
    convt_kernel<<<(int)((tot + TB - 1) / TB), TB, 0, stream>>>(
        cin, Wt[6], Bs[6], hbuf, MIDC, 2, 129, 255, 2, 2, 0);
    extract_weights_pad_kernel<<<gBPNN, TB, 0, stream>>>(hbuf, wRe, wIm);
  }

  // ---- x working copy ----
  hipMemcpyAsync(xA, x0, BNN * sizeof(float), hipMemcpyDeviceToDevice, stream);
  float* cur = xA;
  float* nxt = xB;

  float roots[NITERS];
  for (int i = 0; i < NITERS; i++)
    roots[i] = cosf((float)M_PI * (2 * i + 1) / (2.0f * NITERS));

  unsigned cnt = 0;
  for (int K = 0; K < KOUT; K++) {
    // ---- Chebyshev smoother ----
    init_u_kernel<<<gBNN, TB, 0, stream>>>(ubuf, 1u + cnt++);
    for (int it = 0; it < 20; it++) {
      conv3x3_kernel<<<gBNN, TB, 0, stream>>>(ubuf, kA, ybuf);
      rowmax_abs_kernel<<<NB, TB, 0, stream>>>(ybuf, mbuf);
      scale_kernel<<<gBNN, TB, 0, stream>>>(ybuf, mbuf, ubuf);
    }
    for (int k = 0; k < NITERS; k++) {
      cheby_step_kernel<<<gBNN, TB, 0, stream>>>(cur, nxt, f, kA, mbuf, roots[k]);
      float* tmp = cur; cur = nxt; nxt = tmp;
    }

    // ---- residual into padded spectral layout ----
    residual_pad_kernel<<<gBPNN, TB, 0, stream>>>(cur, f, kA, rp);

    // ---- fft2(r): row pass (real A), then column pass ----
    cgemm256_wmma_kernel<false, true><<<512, 256, 0, stream>>>(
        rp, nullptr, (long)PNN, Fre, Fim, 0L, t1Re, t1Im);     // T1 = r * F
    cgemm256_wmma_kernel<true, true><<<512, 256, 0, stream>>>(
        Fre, Fim, 0L, t1Re, t1Im, (long)PNN, t2Re, t2Im);      // rhat = F * T1

    // ---- out_ft = rhat .* W, rfft truncation + Hermitian extension ----
    spec_mul_herm_kernel<<<gBPNN, TB, 0, stream>>>(t2Re, t2Im, wRe, wIm, sRe, sIm);

    // ---- irfft2: row pass, then column pass (real part only) ----
    cgemm256_wmma_kernel<true, true><<<512, 256, 0, stream>>>(
        sRe, sIm, (long)PNN, FiRe, FiIm, 0L, t1Re, t1Im);      // T = S * Finv
    cgemm256_wmma_kernel<true, false><<<512, 256, 0, stream>>>(
        FiRe, FiIm, 0L, t1Re, t1Im, (long)PNN, ep, nullptr);   // e = Re(Finv*T)

    // ---- x += e ----
    add_from_pad_kernel<<<gBNN, TB, 0, stream>>>(cur, ep);
  }

  // ---- final residual norm ----
  residual_kernel<<<gBNN, TB, 0, stream>>>(cur, f, kA, rbuf);
  zero1_kernel<<<1, 1, 0, stream>>>(nacc);
  sumsq_kernel<<<256, TB, 0, stream>>>(rbuf, nacc);
  finish_norm_kernel<<<1, 1, 0, stream>>>(nacc, (float*)d_out);
}